// BaseTransform_78838419685896
// MI455X (gfx1250) — compile-verified
//
#include <hip/hip_runtime.h>
#include <stdint.h>

namespace {

constexpr int kNCam       = 6;
constexpr int kD          = 118;
constexpr int kFH         = 32;
constexpr int kFW         = 88;
constexpr int kC          = 80;
constexpr int kNX         = 360;
constexpr int kNY         = 360;
constexpr int kPts        = kNCam * kD * kFH * kFW;   // 1993728 (divisible by 16)
constexpr int kPtsPerCam  = kD * kFH * kFW;           // 332288
constexpr int kPtsPerDep  = kFH * kFW;                // 2816
constexpr int kNVox       = kNX * kNY;                // 129600
constexpr long kOutN      = (long)kC * kNVox;         // 10368000
constexpr int kTiles      = kPts / 16;                // 124608
constexpr int kWavesPerBlk = 8;

typedef float v2f __attribute__((ext_vector_type(2)));
typedef float v4f __attribute__((ext_vector_type(4)));
typedef float v8f __attribute__((ext_vector_type(8)));

// ---------------------------------------------------------------- setup ----

__device__ __forceinline__ void inv3(const float m[9], float o[9]) {
  float a=m[0],b=m[1],c=m[2],d=m[3],e=m[4],f=m[5],g=m[6],h=m[7],i=m[8];
  float A =  (e*i - f*h);
  float B = -(d*i - f*g);
  float C =  (d*h - e*g);
  float det = a*A + b*B + c*C;
  float r = 1.0f / det;
  o[0]=A*r;          o[1]=-(b*i-c*h)*r; o[2]= (b*f-c*e)*r;
  o[3]=B*r;          o[4]= (a*i-c*g)*r; o[5]=-(a*f-c*d)*r;
  o[6]=C*r;          o[7]=-(a*h-b*g)*r; o[8]= (a*e-b*d)*r;
}

__device__ __forceinline__ void mm3(const float a[9], const float b[9], float o[9]) {
  for (int r = 0; r < 3; ++r)
    for (int c = 0; c < 3; ++c)
      o[r*3+c] = a[r*3+0]*b[c] + a[r*3+1]*b[3+c] + a[r*3+2]*b[6+c];
}

__device__ __forceinline__ void mv3(const float a[9], const float v[3], float o[3]) {
  for (int r = 0; r < 3; ++r)
    o[r] = a[r*3+0]*v[0] + a[r*3+1]*v[1] + a[r*3+2]*v[2];
}

// xf layout per camera (24 floats):
//   [0..8]   inv(post_rot)          [9..11]  post_trans
//   [12..20] Av = diag(1/dx) * (er*inv(l2e_r)*rot*inv(K))        (row-major)
//   [21..23] tvv = (er*inv(l2e_r)*(trans-l2e_t) + et + off) / dx
// i.e. the voxelization affine ( (geom - (bx-dx/2)) / dx ) is folded into the
// final transform, so the WMMA emits voxel-space coordinates directly.
__global__ void setup_transforms(const float* __restrict__ c2e,
                                 const float* __restrict__ l2e,
                                 const float* __restrict__ intr,
                                 const float* __restrict__ aug,
                                 const float* __restrict__ laug,
                                 float* __restrict__ xf) {
  int t = threadIdx.x;
  if (t >= kNCam) return;
  const float* C2E = c2e  + t * 16;
  const float* KIN = intr + t * 16;
  const float* AUG = aug  + t * 16;
  const float* L2E = l2e;            // B == 1
  const float* LAU = laug;

  float rot[9], in3[9], prot[9], ler[9], er[9];
  float trans[3], ptr3[3], let[3], et[3];
  for (int r = 0; r < 3; ++r) {
    for (int c = 0; c < 3; ++c) {
      rot[r*3+c]  = C2E[r*4+c];
      in3[r*3+c]  = KIN[r*4+c];
      prot[r*3+c] = AUG[r*4+c];
      ler[r*3+c]  = L2E[r*4+c];
      er[r*3+c]   = LAU[r*4+c];
    }
    trans[r] = C2E[r*4+3];
    ptr3[r]  = AUG[r*4+3];
    let[r]   = L2E[r*4+3];
    et[r]    = LAU[r*4+3];
  }

  float Minv[9], Kinv[9], Linv[9], comb[9], t1[9], A[9];
  inv3(prot, Minv);
  inv3(in3,  Kinv);
  inv3(ler,  Linv);
  mm3(rot, Kinv, comb);   // rot * inv(K)
  mm3(Linv, comb, t1);    // inv(l2e_r) * ...
  mm3(er, t1, A);         // er * ...

  float tmv[3], t2[3], tv[3];
  for (int r = 0; r < 3; ++r) tmv[r] = trans[r] - let[r];
  mv3(Linv, tmv, t2);
  mv3(er, t2, tv);
  for (int r = 0; r < 3; ++r) tv[r] += et[r];

  // fold voxelization (offset + divide) into the transform, in double
  const double dxr[3]  = {0.30000001192092896, 0.30000001192092896, 20.0};
  const double offr[3] = {54.0, 54.0, 10.0};

  float* o = xf + t * 24;
  for (int i = 0; i < 9; ++i) o[i]     = Minv[i];
  for (int i = 0; i < 3; ++i) o[9 + i] = ptr3[i];
  for (int r = 0; r < 3; ++r) {
    for (int c = 0; c < 3; ++c)
      o[12 + r*3 + c] = (float)((double)A[r*3+c] / dxr[r]);
    o[21 + r] = (float)(((double)tv[r] + offr[r]) / dxr[r]);
  }
}

// ----------------------------------------------------------------- zero ----

__global__ void fill_zero4(v4f* __restrict__ p, long n4) {
  long i = (long)blockIdx.x * blockDim.x + threadIdx.x;
  long stride = (long)gridDim.x * blockDim.x;
  v4f z = {0.f, 0.f, 0.f, 0.f};
  for (; i < n4; i += stride) p[i] = z;
}

// -------------------------------------------------------------- scatter ----
// One wave per 16-point tile. Both half-waves compute the unprojected point
// p2 for point m = lane&15; the WMMA A-matrix layout (lanes 0-15: K0/K1,
// lanes 16-31: K2/K3 of the same rows) then needs no cross-lane traffic.
// B holds the fused (voxel-scaled) transform columns, so D[m][0..2] are the
// voxel-space coords of point m; extracted wave-uniform via v_readlane.
// VS/CS are compile-time accumulator strides (voxel / channel).

template <int VS, int CS>
__global__ void __launch_bounds__(256)
bev_scatter(const float* __restrict__ feats, const float* __restrict__ xf,
            float* __restrict__ acc) {
  const int lane = threadIdx.x & 31;
  const int wave = threadIdx.x >> 5;
  const int tile = blockIdx.x * kWavesPerBlk + wave;
  const int base = tile << 4;
  const int m    = lane & 15;
  const bool hi  = lane >= 16;

  // prefetch this tile's 5KB feature block (overlaps the geometry preamble)
  const float* fbase = feats + (size_t)base * kC;
  __builtin_prefetch(fbase + lane * 32, 0, 1);
  __builtin_prefetch(fbase + 1024 + lane * 32, 0, 1);

  const unsigned p = (unsigned)(base + m);
  unsigned n    = p / (unsigned)kPtsPerCam;
  unsigned rem  = p - n * kPtsPerCam;
  unsigned dep  = rem / (unsigned)kPtsPerDep;
  unsigned rem2 = rem - dep * kPtsPerDep;
  unsigned h    = rem2 / (unsigned)kFW;
  unsigned w    = rem2 - h * kFW;

  const float fx = (float)w * (703.0f / 87.0f);   // linspace(0, 703, 88)
  const float fy = (float)h * (255.0f / 31.0f);   // linspace(0, 255, 32)
  const float fd = 1.0f + 0.5f * (float)dep;      // arange(1, 60, 0.5)

  const float* T = xf + n * 24;
  const float ux = fx - T[9], uy = fy - T[10], ud = fd - T[11];
  const float p1x = T[0]*ux + T[1]*uy + T[2]*ud;
  const float p1y = T[3]*ux + T[4]*uy + T[5]*ud;
  const float p1z = T[6]*ux + T[7]*uy + T[8]*ud;

  // A-matrix (16x4): row m = (p2x, p2y, p2z, 1) after perspective-unproject
  v2f Av;
  Av.x = hi ? p1z : p1x * p1z;
  Av.y = hi ? 1.0f : p1y * p1z;

  // B-matrix (4x16): column n<3 = fused transform row n (cols 3..15 zero)
  const int  nc   = (m < 3) ? m : 0;
  const bool vcol = (m < 3);
  const float g0 = T[12 + nc*3 + 0];
  const float g1 = T[12 + nc*3 + 1];
  const float g2 = T[12 + nc*3 + 2];
  const float g3 = T[21 + nc];
  v2f Bv;
  Bv.x = vcol ? (hi ? g2 : g0) : 0.0f;
  Bv.y = vcol ? (hi ? g3 : g1) : 0.0f;

  v8f Cz = {0.f, 0.f, 0.f, 0.f, 0.f, 0.f, 0.f, 0.f};
  v8f Dm = __builtin_amdgcn_wmma_f32_16x16x4_f32(
      /*neg_a=*/false, Av, /*neg_b=*/false, Bv,
      /*c_mod=*/(short)0, Cz, /*reuse_a=*/false, /*reuse_b=*/false);

  #pragma unroll
  for (int q = 0; q < 16; ++q) {
    const int r  = q & 7;
    const int sl = (q & 8) ? 16 : 0;
    const float gx = __builtin_bit_cast(float,
        __builtin_amdgcn_readlane(__builtin_bit_cast(int, Dm[r]), sl + 0));
    const float gy = __builtin_bit_cast(float,
        __builtin_amdgcn_readlane(__builtin_bit_cast(int, Dm[r]), sl + 1));
    const float gz = __builtin_bit_cast(float,
        __builtin_amdgcn_readlane(__builtin_bit_cast(int, Dm[r]), sl + 2));

    // truncation toward zero matches the reference's astype(int32)
    const int cx = (int)gx;
    const int cy = (int)gy;
    const int cz = (int)gz;
    if ((unsigned)cx < (unsigned)kNX && (unsigned)cy < (unsigned)kNY && cz == 0) {
      const int vox = cx * kNY + cy;
      if (lane < kC / 4) {   // 20 lanes x float4 = 80 channels
        v4f f = __builtin_nontemporal_load(
            ((const v4f*)(fbase + q * kC)) + lane);
        float* dst = acc + (size_t)vox * VS + (size_t)(lane * 4) * CS;
        atomicAdd(dst + 0 * CS, f.x);
        atomicAdd(dst + 1 * CS, f.y);
        atomicAdd(dst + 2 * CS, f.z);
        atomicAdd(dst + 3 * CS, f.w);
      }
    }
  }
}

// ------------------------------------------------------------ transpose ----
// acc[s*80 + c]  ->  out[c*129600 + s], 32-spatial x 80-channel LDS tiles,
// padded to 81 floats/row so the strided read phase is bank-conflict free.

__global__ void __launch_bounds__(256)
bev_transpose(const float* __restrict__ acc, float* __restrict__ out) {
  __shared__ float tile[32 * 81];
  const int s0  = blockIdx.x * 32;
  const int tid = threadIdx.x;
  const float* src = acc + (size_t)s0 * kC;
  for (int i = tid; i < 32 * kC; i += 256) {
    int sl = i / kC;
    int c  = i - sl * kC;
    tile[sl * 81 + c] = src[i];
  }
  __syncthreads();
  for (int i = tid; i < 32 * kC; i += 256) {
    int c  = i >> 5;
    int sl = i & 31;
    out[(size_t)c * kNVox + s0 + sl] = tile[sl * 81 + c];
  }
}

} // namespace

// ----------------------------------------------------------------- host ----

extern "C" void kernel_launch(void* const* d_in, const int* in_sizes, int n_in,
                              void* d_out, int out_size, void* d_ws, size_t ws_size,
                              hipStream_t stream) {
  const float* feats = (const float*)d_in[0];
  const float* c2e   = (const float*)d_in[1];
  const float* l2e   = (const float*)d_in[2];
  const float* intr  = (const float*)d_in[3];
  const float* aug   = (const float*)d_in[4];
  const float* laug  = (const float*)d_in[5];
  float* out = (float*)d_out;

  float* xf = (float*)d_ws;                       // 576 B of fused transforms
  const size_t accOff   = 1024;
  const size_t accBytes = (size_t)kOutN * sizeof(float);
  const bool   staged   = (ws_size >= accOff + accBytes);

  setup_transforms<<<1, 32, 0, stream>>>(c2e, l2e, intr, aug, laug, xf);

  if (staged) {
    float* acc = (float*)((char*)d_ws + accOff);
    fill_zero4<<<2048, 256, 0, stream>>>((v4f*)acc, kOutN / 4);
    bev_scatter<kC, 1><<<kTiles / kWavesPerBlk, 256, 0, stream>>>(feats, xf, acc);
    bev_transpose<<<kNVox / 32, 256, 0, stream>>>(acc, out);
  } else {
    // workspace too small: accumulate directly in output layout
    fill_zero4<<<2048, 256, 0, stream>>>((v4f*)out, kOutN / 4);
    bev_scatter<1, kNVox><<<kTiles / kWavesPerBlk, 256, 0, stream>>>(feats, xf, out);
  }
}